// SynthesisNet_83047487635951
// MI455X (gfx1250) — compile-verified
//
#include <hip/hip_runtime.h>
#include <hip/hip_bf16.h>
#include <math.h>

#define BB 4096
#define SS 512
#define USE_ASYNC 1
#define TILE_F (16 * 68)  // 16 rows, stride 68 -> conflict-free column reads

typedef __attribute__((ext_vector_type(2))) float v2f;
typedef __attribute__((ext_vector_type(8))) float v8f;

__device__ __forceinline__ v8f wmma_f32_k4(v2f a, v2f b, v8f c) {
  // D = A(16x4 f32) x B(4x16 f32) + C(16x16 f32), full fp32 precision
  return __builtin_amdgcn_wmma_f32_16x16x4_f32(false, a, false, b, (short)0, c,
                                               false, false);
}

__device__ __forceinline__ float sigmoidf_(float x) {
  return 1.0f / (1.0f + expf(-x));
}

#if USE_ASYNC
#define WAIT_ASYNC() asm volatile("s_wait_asynccnt 0x0" ::: "memory")
#else
#define WAIT_ASYNC()
#endif

// ------------------------------------------------- swizzle+pad weights
// dst[((p*Cd)+c)*2+e] = src[(2p+e)*Cs + c]  (zero beyond R/Cs)
// i.e. row-pairs interleaved so a WMMA B-fragment {W[k][n],W[k+1][n]} is one
// contiguous 8-byte load.
__global__ void swz_pad_kernel(const float* __restrict__ src,
                               float* __restrict__ dst, int R, int Cs, int Rp,
                               int Cd) {
  int idx = blockIdx.x * blockDim.x + threadIdx.x;
  if (idx >= Rp * Cd) return;
  int e = idx & 1;
  int t = idx >> 1;
  int c = t % Cd;
  int p = t / Cd;
  int r = 2 * p + e;
  dst[idx] = (r < R && c < Cs) ? src[(size_t)r * Cs + c] : 0.0f;
}

// ------------------------------------------------------------- concatenate
// dst row layout: [x0 (3) | mid (midw, optional) | w (80) | zero pad]
__global__ void concat_kernel(const float* __restrict__ x,
                              const float* __restrict__ mid, int midw,
                              const float* __restrict__ wv,
                              float* __restrict__ dst, int Kp) {
  int idx = blockIdx.x * blockDim.x + threadIdx.x;
  if (idx >= BB * Kp) return;
  int b = idx / Kp, c = idx % Kp;
  float v = 0.0f;
  if (c < 3)
    v = x[b * 3 + c];
  else if (c < 3 + midw)
    v = mid[(size_t)b * midw + (c - 3)];
  else if (c < 3 + midw + 80)
    v = wv[b * 80 + (c - 3 - midw)];
  dst[idx] = v;
}

// ---------------------------------------------- async A-chunk staging (LDS)
__device__ __forceinline__ void stage_chunk(float* dst /* shared */,
                                            const float* __restrict__ src,
                                            int stride, int m0, int kc) {
  const int t = threadIdx.x;
  const int srow = t >> 4;         // 0..15
  const int scol = (t & 15) << 2;  // float4 column group
  const float* gp = src + (size_t)(m0 + srow) * stride + kc + scol;
  float* lp = dst + srow * 68 + scol;
#if USE_ASYNC
  unsigned ldsoff = (unsigned)(uintptr_t)lp;  // low 32 bits = LDS offset
  asm volatile("global_load_async_to_lds_b128 %0, %1, off" ::"v"(ldsoff),
               "v"(gp)
               : "memory");
#else
  *(float4*)lp = *(const float4*)gp;
#endif
}

// -------------------------------------------------- GEMM phase (one input)
// Double-buffered async LDS staging of 16x64 A-chunks; swizzled B so each
// fragment is one b64 load.  K must be a multiple of 64.
__device__ __forceinline__ void ff_phase(
    float* tile /* __shared__ [2][TILE_F] */, const float* __restrict__ src,
    int stride, int K, const float* __restrict__ Wj,
    const float* __restrict__ Wk, int m0, int n0, int lane, v8f* accj,
    v8f* acck) {
  const int lm = lane & 15;
  const int half = lane >> 4;
  const int nchunks = K >> 6;

  stage_chunk(tile, src, stride, m0, 0);
  for (int i = 0; i < nchunks; ++i) {
    float* cur = tile + (i & 1) * TILE_F;
    float* nxt = tile + ((i & 1) ^ 1) * TILE_F;
    WAIT_ASYNC();       // chunk i landed in LDS (per wave)
    __syncthreads();    // visible to all; all waves done reading buffer `nxt`
    if (i + 1 < nchunks) stage_chunk(nxt, src, stride, m0, (i + 1) << 6);

    const int kc = i << 6;
#pragma unroll 4
    for (int kk = 0; kk < 64; kk += 4) {
      v2f a;
      a.x = cur[lm * 68 + kk + half * 2];
      a.y = cur[lm * 68 + kk + half * 2 + 1];
      const int kp = (kc + kk + half * 2) >> 1;  // swizzled row-pair index
#pragma unroll
      for (int nt = 0; nt < 4; ++nt) {
        const int n = n0 + nt * 16 + lm;
        v2f bj = *(const v2f*)(Wj + ((size_t)kp * SS + n) * 2);
        v2f bk = *(const v2f*)(Wk + ((size_t)kp * SS + n) * 2);
        accj[nt] = wmma_f32_k4(a, bj, accj[nt]);
        acck[nt] = wmma_f32_k4(a, bk, acck[nt]);
      }
    }
  }
  __syncthreads();  // phase end: next phase's prologue stage is safe
}

// ------------------------------------------------- dual-gate WMMA ff GEMM
// Block: 256 threads = 8 waves; block computes 16 rows x 512 cols.
// Wave w handles the 16x64 strip at n0 = w*64 for BOTH gates:
//   j = sigmoid(X@Wjx + bjx + H@Wjh + bjh)
//   k = sigmoid(X@Wkx + bkx + H@Wkh + bkh)
//   h_out = j*(1-h_prev) + (1-k)*h_prev
__global__ __launch_bounds__(256) void ff_gemm_kernel(
    const float* __restrict__ X, int Kxp, const float* __restrict__ Wjx,
    const float* __restrict__ Wkx, const float* __restrict__ H,
    const float* __restrict__ Wjh, const float* __restrict__ Wkh,
    const float* __restrict__ bjx, const float* __restrict__ bjh,
    const float* __restrict__ bkx, const float* __restrict__ bkh,
    const float* __restrict__ h_prev, float* __restrict__ h_out) {
  __shared__ float tile[2 * TILE_F];
  const int lane = threadIdx.x & 31;
  const int wave = threadIdx.x >> 5;
  const int m0 = blockIdx.x * 16;
  const int n0 = wave * 64;

  v8f accj[4] = {};
  v8f acck[4] = {};

  ff_phase(tile, X, Kxp, Kxp, Wjx, Wkx, m0, n0, lane, accj, acck);
  ff_phase(tile, H, SS, SS, Wjh, Wkh, m0, n0, lane, accj, acck);

  const int lm = lane & 15;
  const int half = lane >> 4;
#pragma unroll
  for (int nt = 0; nt < 4; ++nt) {
    const int n = n0 + nt * 16 + lm;
    const float jb = bjx[n] + bjh[n];
    const float kb = bkx[n] + bkh[n];
#pragma unroll
    for (int r = 0; r < 8; ++r) {
      const int m = m0 + r + half * 8;
      const float hp = h_prev[(size_t)m * SS + n];
      const float j = sigmoidf_(accj[nt][r] + jb);
      const float kk = sigmoidf_(acck[nt][r] + kb);
      h_out[(size_t)m * SS + n] = j * (1.0f - hp) + (1.0f - kk) * hp;
    }
  }
}

// --------------------------------------------------- plain WMMA GEMM (out)
// C(B x 128) = A(B x 512) @ Wswz(512 x 128, row-pair swizzled)
__global__ __launch_bounds__(32) void out_gemm_kernel(
    const float* __restrict__ A, const float* __restrict__ W,
    float* __restrict__ C) {
  const int lane = threadIdx.x;
  const int half = lane >> 4;
  const int lm = lane & 15;
  const int m0 = blockIdx.x * 16;
  const int n0 = blockIdx.y * 16;
  const int N = 128;

  v8f acc = {};
  const float* arow = A + (size_t)(m0 + lm) * SS;
  for (int k = 0; k < SS; k += 4) {
    const int ka = k + half * 2;
    v2f a;
    a.x = arow[ka];
    a.y = arow[ka + 1];
    v2f b = *(const v2f*)(W + ((size_t)(ka >> 1) * N + (n0 + lm)) * 2);
    acc = wmma_f32_k4(a, b, acc);
  }
#pragma unroll
  for (int r = 0; r < 8; ++r) {
    const int m = m0 + r + half * 8;
    C[(size_t)m * N + (n0 + lm)] = acc[r];
  }
}

// -------------------------------------------------------------- kg = h1@Ww
__global__ __launch_bounds__(32) void kg_kernel(const float* __restrict__ h1,
                                                const float* __restrict__ Ww,
                                                const float* __restrict__ bw,
                                                float* __restrict__ kg) {
  const int b = blockIdx.x;
  const int t = threadIdx.x;
  if (t < 30) {
    float s = bw[t];
    const float* hr = h1 + (size_t)b * SS;
    for (int i = 0; i < SS; ++i) s += hr[i] * Ww[i * 30 + t];
    kg[b * 32 + t] = s;
  }
}

// ------------------------------------------------------- attention window
__global__ __launch_bounds__(256) void attention_kernel(
    const float* __restrict__ kg, const float* __restrict__ init_kappa,
    const float* __restrict__ c_vec, float* __restrict__ w_out,
    float* __restrict__ phi_out, float* __restrict__ kappa_out) {
  __shared__ float s_abk[32];  // [0..9] alpha, [10..19] beta, [20..29] kappa
  __shared__ float s_phi[256];
  const int b = blockIdx.x;
  const int t = threadIdx.x;

  if (t < 30) {
    float e = expf(kg[b * 32 + t]);
    if (t >= 20) {
      e += init_kappa[b * 10 + (t - 20)];
      kappa_out[b * 10 + (t - 20)] = e;
    }
    s_abk[t] = e;
  }
  __syncthreads();

  {  // phi[u] = sum_k exp(-beta_k * (kappa_k - u)^2) * alpha_k
    const float u = (float)t;
    float phi = 0.0f;
#pragma unroll
    for (int i = 0; i < 10; ++i) {
      const float d = s_abk[20 + i] - u;
      phi += expf(-s_abk[10 + i] * d * d) * s_abk[i];
    }
    s_phi[t] = phi;
    phi_out[(size_t)b * 256 + t] = phi;
  }
  __syncthreads();

  if (t < 80) {  // w[c] = sum_u phi[u] * c_vec[b,u,c]  (coalesced over c)
    const float* cb = c_vec + (size_t)b * 256 * 80 + t;
    float s = 0.0f;
    for (int u = 0; u < 256; ++u) s += s_phi[u] * cb[(size_t)u * 80];
    w_out[b * 80 + t] = s;
  }
}

// ------------------------------------------------------------ postprocess
__global__ __launch_bounds__(128) void postproc_kernel(
    const float* __restrict__ out_pre, const float* __restrict__ bl,
    float* __restrict__ eos, float* __restrict__ pi, float* __restrict__ mu1,
    float* __restrict__ mu2, float* __restrict__ s1, float* __restrict__ s2,
    float* __restrict__ rho) {
  __shared__ float s_o[121];
  __shared__ float s_max, s_sum;
  const int b = blockIdx.x;
  const int t = threadIdx.x;
  if (t < 121) s_o[t] = out_pre[(size_t)b * 128 + t] + bl[t];
  __syncthreads();

  if (t == 0) {
    eos[b] = sigmoidf_(-s_o[0]);
    float m = s_o[1];
    for (int i = 2; i <= 20; ++i) m = fmaxf(m, s_o[i]);
    float sum = 0.0f;
    for (int i = 1; i <= 20; ++i) sum += expf(s_o[i] - m);
    s_max = m;
    s_sum = sum;
  }
  __syncthreads();

  if (t < 20) {
    pi[b * 20 + t] = expf(s_o[1 + t] - s_max) / s_sum;
    mu1[b * 20 + t] = s_o[21 + t];
    mu2[b * 20 + t] = s_o[41 + t];
    s1[b * 20 + t] = expf(s_o[61 + t]);
    s2[b * 20 + t] = expf(s_o[81 + t]);
    rho[b * 20 + t] = tanhf(s_o[101 + t]);
  }
}

// ---------------------------------------------------------------- launch
extern "C" void kernel_launch(void* const* d_in, const int* in_sizes, int n_in,
                              void* d_out, int out_size, void* d_ws,
                              size_t ws_size, hipStream_t stream) {
  const int B = BB;
  const int K1P = 128;  // 83 padded to multiple of 64
  const int K2P = 640;  // 595 padded to multiple of 64

  const float* x = (const float*)d_in[0];           // (B,1,3)
  const float* c_vec = (const float*)d_in[1];       // (B,256,80)
  const float* rnn1_h = (const float*)d_in[2];      // (B,512)
  const float* rnn2_h = (const float*)d_in[3];
  const float* rnn3_h = (const float*)d_in[4];
  const float* init_w = (const float*)d_in[5];      // (B,1,80)
  const float* init_kappa = (const float*)d_in[6];  // (B,10,1)
  const float* r1[8];
  const float* r2[8];
  const float* r3[8];
  for (int i = 0; i < 8; ++i) {  // Wjx,bjx,Wjh,bjh,Wkx,bkx,Wkh,bkh
    r1[i] = (const float*)d_in[7 + i];
    r2[i] = (const float*)d_in[15 + i];
    r3[i] = (const float*)d_in[23 + i];
  }
  const float* Ww = (const float*)d_in[31];  // (512,30)
  const float* bw = (const float*)d_in[32];
  const float* Wl = (const float*)d_in[33];  // (512,121)
  const float* bl = (const float*)d_in[34];

  float* out = (float*)d_out;
  float* o_eos = out;
  float* o_pi = out + (size_t)B;
  float* o_mu1 = out + (size_t)B * 21;
  float* o_mu2 = out + (size_t)B * 41;
  float* o_s1 = out + (size_t)B * 61;
  float* o_s2 = out + (size_t)B * 81;
  float* o_rho = out + (size_t)B * 101;
  float* o_h1 = out + (size_t)B * 121;
  float* o_h2 = out + (size_t)B * 633;
  float* o_h3 = out + (size_t)B * 1145;
  float* o_w = out + (size_t)B * 1657;
  float* o_phi = out + (size_t)B * 1737;
  float* o_kappa = out + (size_t)B * 1993;

  float* ws = (float*)d_ws;
  size_t off = 0;
  float* wsX1 = ws + off; off += (size_t)B * K1P;
  float* wsX2 = ws + off; off += (size_t)B * K2P;
  float* wsX3 = ws + off; off += (size_t)B * K2P;
  float* wsW1j = ws + off;  off += (size_t)K1P * 512;
  float* wsW1k = ws + off;  off += (size_t)K1P * 512;
  float* wsW1jh = ws + off; off += (size_t)512 * 512;
  float* wsW1kh = ws + off; off += (size_t)512 * 512;
  float* wsW2j = ws + off;  off += (size_t)K2P * 512;
  float* wsW2k = ws + off;  off += (size_t)K2P * 512;
  float* wsW2jh = ws + off; off += (size_t)512 * 512;
  float* wsW2kh = ws + off; off += (size_t)512 * 512;
  float* wsW3j = ws + off;  off += (size_t)K2P * 512;
  float* wsW3k = ws + off;  off += (size_t)K2P * 512;
  float* wsW3jh = ws + off; off += (size_t)512 * 512;
  float* wsW3kh = ws + off; off += (size_t)512 * 512;
  float* wsWl = ws + off;   off += 512 * 128;
  float* wsKg = ws + off;   off += (size_t)B * 32;
  float* wsOutPre = ws + off; off += (size_t)B * 128;

  const int TPB = 256;
  auto nb = [](size_t n, int t) { return (int)((n + t - 1) / t); };

  // --- swizzle (row-pair interleave) + zero-pad all weights into ws
  swz_pad_kernel<<<nb((size_t)K1P * 512, TPB), TPB, 0, stream>>>(r1[0], wsW1j, 83, 512, K1P, 512);
  swz_pad_kernel<<<nb((size_t)K1P * 512, TPB), TPB, 0, stream>>>(r1[4], wsW1k, 83, 512, K1P, 512);
  swz_pad_kernel<<<nb((size_t)512 * 512, TPB), TPB, 0, stream>>>(r1[2], wsW1jh, 512, 512, 512, 512);
  swz_pad_kernel<<<nb((size_t)512 * 512, TPB), TPB, 0, stream>>>(r1[6], wsW1kh, 512, 512, 512, 512);
  swz_pad_kernel<<<nb((size_t)K2P * 512, TPB), TPB, 0, stream>>>(r2[0], wsW2j, 595, 512, K2P, 512);
  swz_pad_kernel<<<nb((size_t)K2P * 512, TPB), TPB, 0, stream>>>(r2[4], wsW2k, 595, 512, K2P, 512);
  swz_pad_kernel<<<nb((size_t)512 * 512, TPB), TPB, 0, stream>>>(r2[2], wsW2jh, 512, 512, 512, 512);
  swz_pad_kernel<<<nb((size_t)512 * 512, TPB), TPB, 0, stream>>>(r2[6], wsW2kh, 512, 512, 512, 512);
  swz_pad_kernel<<<nb((size_t)K2P * 512, TPB), TPB, 0, stream>>>(r3[0], wsW3j, 595, 512, K2P, 512);
  swz_pad_kernel<<<nb((size_t)K2P * 512, TPB), TPB, 0, stream>>>(r3[4], wsW3k, 595, 512, K2P, 512);
  swz_pad_kernel<<<nb((size_t)512 * 512, TPB), TPB, 0, stream>>>(r3[2], wsW3jh, 512, 512, 512, 512);
  swz_pad_kernel<<<nb((size_t)512 * 512, TPB), TPB, 0, stream>>>(r3[6], wsW3kh, 512, 512, 512, 512);
  swz_pad_kernel<<<nb((size_t)512 * 128, TPB), TPB, 0, stream>>>(Wl, wsWl, 512, 121, 512, 128);

  // --- rnn1: h1 = ff([x0|w], rnn1_h)
  concat_kernel<<<nb((size_t)B * K1P, TPB), TPB, 0, stream>>>(x, nullptr, 0, init_w, wsX1, K1P);
  ff_gemm_kernel<<<B / 16, 256, 0, stream>>>(wsX1, K1P, wsW1j, wsW1k, rnn1_h,
                                             wsW1jh, wsW1kh, r1[1], r1[3],
                                             r1[5], r1[7], rnn1_h, o_h1);

  // --- attention: kg -> alpha/beta/kappa -> phi -> w
  kg_kernel<<<B, 32, 0, stream>>>(o_h1, Ww, bw, wsKg);
  attention_kernel<<<B, 256, 0, stream>>>(wsKg, init_kappa, c_vec, o_w, o_phi,
                                          o_kappa);

  // --- rnn2: h2 = ff([x0|h1|w], rnn2_h)
  concat_kernel<<<nb((size_t)B * K2P, TPB), TPB, 0, stream>>>(x, o_h1, 512, o_w, wsX2, K2P);
  ff_gemm_kernel<<<B / 16, 256, 0, stream>>>(wsX2, K2P, wsW2j, wsW2k, rnn2_h,
                                             wsW2jh, wsW2kh, r2[1], r2[3],
                                             r2[5], r2[7], rnn2_h, o_h2);

  // --- rnn3: h3 = ff([x0|h2|w], rnn3_h)
  concat_kernel<<<nb((size_t)B * K2P, TPB), TPB, 0, stream>>>(x, o_h2, 512, o_w, wsX3, K2P);
  ff_gemm_kernel<<<B / 16, 256, 0, stream>>>(wsX3, K2P, wsW3j, wsW3k, rnn3_h,
                                             wsW3jh, wsW3kh, r3[1], r3[3],
                                             r3[5], r3[7], rnn3_h, o_h3);

  // --- output head
  {
    dim3 grid(B / 16, 128 / 16);
    out_gemm_kernel<<<grid, 32, 0, stream>>>(o_h3, wsWl, wsOutPre);
  }
  postproc_kernel<<<B, 128, 0, stream>>>(wsOutPre, bl, o_eos, o_pi, o_mu1,
                                         o_mu2, o_s1, o_s2, o_rho);
}